// RegionProposalNetwork_48163763258065
// MI455X (gfx1250) — compile-verified
//
#include <hip/hip_runtime.h>
#include <hip/hip_bf16.h>
#include <math.h>

// ---------------- static problem configuration ----------------
#define A_TOTAL   159882
#define BATCH     4
#define K_SEL     4507           // 1000+1000+1000+1000+507 selected per image
#define K_PAD     4608           // padded per-image stride
#define PAD_N     8192           // bitonic sort size
#define NMS_WORDS 71             // ceil(4507/64)
#define NMS_ROWS  4544           // 71*64
#define POST_N    1000
#define IMG_WF    800.0f
#define IMG_HF    800.0f
#define NMS_TH    0.7f
#define MIN_SZ    0.001f
#define BBOX_CLIP 4.135166556742356f   // log(1000/16)

__constant__ int c_lvl_n[5]   = {120000, 30000, 7500, 1875, 507};
__constant__ int c_lvl_off[5] = {0, 120000, 150000, 157500, 159375};
__constant__ int c_lvl_k[5]   = {1000, 1000, 1000, 1000, 507};
__constant__ int c_sel_off[5] = {0, 1000, 2000, 3000, 4000};

typedef unsigned int       u32;
typedef unsigned long long u64;

typedef unsigned int u32x4 __attribute__((ext_vector_type(4)));
typedef int          i32x4 __attribute__((ext_vector_type(4)));
typedef int          i32x8 __attribute__((ext_vector_type(8)));

// TDM builtin arity differs: ROCm7.2/clang-22 = 5 args, amdgpu-toolchain/clang-23 = 6 args.
#if defined(__clang_major__) && (__clang_major__ >= 23)
#define TDM_LOAD_TO_LDS(g0, g1, g2, g3) \
  __builtin_amdgcn_tensor_load_to_lds((g0), (g1), (g2), (g3), (i32x8){0,0,0,0,0,0,0,0}, 0)
#else
#define TDM_LOAD_TO_LDS(g0, g1, g2, g3) \
  __builtin_amdgcn_tensor_load_to_lds((g0), (g1), (g2), (g3), 0)
#endif

// monotone float -> u32 key (larger float => larger key)
__device__ __forceinline__ u32 f2key(float f) {
  u32 u = __float_as_uint(f);
  return (u & 0x80000000u) ? ~u : (u | 0x80000000u);
}

// =====================================================================
// Kernel 1: per-(image,level) top-k via 4096-bucket radix select.
// 20 workgroups (BATCH*5), 1024 threads each.
// =====================================================================
__global__ void __launch_bounds__(1024)
topk_select_kernel(const float* __restrict__ obj,
                   u32* __restrict__ sel_key, int* __restrict__ sel_idx) {
  const int b   = blockIdx.x / 5;
  const int lvl = blockIdx.x % 5;
  const int n   = c_lvl_n[lvl];
  const int off = c_lvl_off[lvl];
  const int k   = c_lvl_k[lvl];
  const int tid = threadIdx.x;
  const float* __restrict__ p = obj + (size_t)b * A_TOTAL + off;

  __shared__ u32 hist[4096];
  __shared__ u32 s_B, s_quota, s_cabove, s_cnt_hi, s_cnt_eq;

  for (int i = tid; i < 4096; i += 1024) hist[i] = 0u;
  __syncthreads();

  for (int i = tid; i < n; i += 1024) {
    __builtin_prefetch(p + i + 8192, 0, 1);     // global_prefetch_b8
    atomicAdd(&hist[f2key(p[i]) >> 20], 1u);
  }
  __syncthreads();

  if (tid == 0) {
    u32 cum = 0; int B = 0;
    for (int bkt = 4095; bkt >= 0; --bkt) {
      u32 h = hist[bkt];
      if (cum + h >= (u32)k) { B = bkt; break; }
      cum += h;
    }
    s_B = (u32)B; s_cabove = cum; s_quota = (u32)k - cum;
    s_cnt_hi = 0u; s_cnt_eq = 0u;
  }
  __syncthreads();

  const u32 B = s_B, quota = s_quota, ca = s_cabove;
  u32* __restrict__ sk = sel_key + (size_t)b * K_PAD + c_sel_off[lvl];
  int* __restrict__ si = sel_idx + (size_t)b * K_PAD + c_sel_off[lvl];

  for (int i = tid; i < n; i += 1024) {
    __builtin_prefetch(p + i + 8192, 0, 1);
    u32 key = f2key(p[i]);
    u32 bkt = key >> 20;
    if (bkt > B) {
      u32 pos = atomicAdd(&s_cnt_hi, 1u);
      sk[pos] = key; si[pos] = off + i;
    } else if (bkt == B) {
      u32 t = atomicAdd(&s_cnt_eq, 1u);
      if (t < quota) { sk[ca + t] = key; si[ca + t] = off + i; }
    }
  }
}

// =====================================================================
// Kernel 2: gather + decode + clip + sigmoid for the selected anchors.
// =====================================================================
__global__ void __launch_bounds__(256)
decode_kernel(const float* __restrict__ obj,
              const float* __restrict__ deltas,
              const float* __restrict__ anchors,
              const int* __restrict__ sel_idx,
              float4* __restrict__ dec_box, float* __restrict__ dec_score,
              u32* __restrict__ dec_lvl, u32* __restrict__ dec_key) {
  int e = blockIdx.x * 256 + threadIdx.x;
  if (e >= BATCH * K_SEL) return;
  int b = e / K_SEL, slot = e % K_SEL;
  int lvl = (slot < 4000) ? (slot / 1000) : 4;
  int si = b * K_PAD + slot;
  int idx = sel_idx[si];

  const float4 a = *(const float4*)(anchors + (size_t)idx * 4);
  const float4 d = *(const float4*)(deltas + ((size_t)b * A_TOTAL + idx) * 4);

  float wa = a.z - a.x, ha = a.w - a.y;
  float cxa = a.x + 0.5f * wa, cya = a.y + 0.5f * ha;
  float dw = fminf(d.z, BBOX_CLIP), dh = fminf(d.w, BBOX_CLIP);
  float cx = d.x * wa + cxa, cy = d.y * ha + cya;
  float w = expf(dw) * wa, h = expf(dh) * ha;

  float x1 = fminf(fmaxf(cx - 0.5f * w, 0.0f), IMG_WF);
  float y1 = fminf(fmaxf(cy - 0.5f * h, 0.0f), IMG_HF);
  float x2 = fminf(fmaxf(cx + 0.5f * w, 0.0f), IMG_WF);
  float y2 = fminf(fmaxf(cy + 0.5f * h, 0.0f), IMG_HF);
  bool valid = (x2 - x1 >= MIN_SZ) && (y2 - y1 >= MIN_SZ);

  float logit = obj[(size_t)b * A_TOTAL + idx];
  float score = 1.0f / (1.0f + expf(-logit));

  dec_box[si]   = make_float4(x1, y1, x2, y2);
  dec_score[si] = score;
  dec_lvl[si]   = (u32)lvl;
  dec_key[si]   = valid ? f2key(logit) : 0u;
}

// =====================================================================
// Kernel 3: per-image bitonic sort (descending) of (key<<32 | slot) in LDS.
// 4 blocks x 1024 threads, 64KB dynamic LDS.
// =====================================================================
__global__ void __launch_bounds__(1024)
sort_kernel(const u32* __restrict__ dec_key, const float4* __restrict__ dec_box,
            const float* __restrict__ dec_score, const u32* __restrict__ dec_lvl,
            float4* __restrict__ s_box, float* __restrict__ s_score,
            u32* __restrict__ s_lv) {
  extern __shared__ u64 items[];   // PAD_N entries
  const int img = blockIdx.x, tid = threadIdx.x;
  const size_t base = (size_t)img * K_PAD;

  for (int i = tid; i < PAD_N; i += 1024) {
    if (i < K_SEL) items[i] = ((u64)dec_key[base + i] << 32) | (u32)i;
    else           items[i] = 0xFFFFFFFFull;       // key=0 pad, slot marker
  }
  __syncthreads();

  for (int k = 2; k <= PAD_N; k <<= 1) {
    for (int j = k >> 1; j > 0; j >>= 1) {
      for (int i = tid; i < PAD_N; i += 1024) {
        int ixj = i ^ j;
        if (ixj > i) {
          bool up = ((i & k) == 0);          // descending overall
          u64 va = items[i], vb = items[ixj];
          if ((va < vb) == up) { items[i] = vb; items[ixj] = va; }
        }
      }
      __syncthreads();
    }
  }

  for (int p = tid; p < K_SEL; p += 1024) {
    u64 it = items[p];
    u32 slot = (u32)(it & 0xFFFFFFFFull);
    u32 key  = (u32)(it >> 32);
    if (key != 0u && slot < K_SEL) {
      s_box[base + p]   = dec_box[base + slot];
      s_score[base + p] = dec_score[base + slot];
      s_lv[base + p]    = dec_lvl[base + slot] | 0x100u;   // bit8 = valid
    } else {
      s_box[base + p]   = make_float4(0.f, 0.f, 0.f, 0.f);
      s_score[base + p] = 0.0f;
      s_lv[base + p]    = 0u;                               // invalid
    }
  }
}

// =====================================================================
// Kernel 4: suppression bitmask, 64x64 tiles (upper triangle).
// grid (j_tile=71, i_tile=71, img=4), 64 threads. The contiguous 1KB
// j-box tile is staged into LDS by the Tensor Data Mover (1-D tile).
// =====================================================================
__global__ void __launch_bounds__(64)
nms_mask_kernel(const float4* __restrict__ s_box, const u32* __restrict__ s_lv,
                u64* __restrict__ mask) {
  const int jt = blockIdx.x, it = blockIdx.y, img = blockIdx.z;
  if (jt < it) return;                    // only j > i can be suppressed
  const int t = threadIdx.x;

  __shared__ float4 jb[64];
  __shared__ u32    jl[64];

  // --- TDM: 1KB contiguous global -> LDS (64 float4 boxes) ---
  if (t == 0) {
    const void* gptr = (const void*)(s_box + (size_t)img * K_PAD + (size_t)jt * 64);
    u64 ga  = (u64)(uintptr_t)gptr;
    u32 lds = (u32)(uintptr_t)&jb[0];
    u32x4 g0;
    g0.x = 1u;                                         // count = 1 valid descriptor
    g0.y = lds;                                        // lds_addr
    g0.z = (u32)(ga & 0xFFFFFFFFull);                  // global_addr[31:0]
    g0.w = (u32)((ga >> 32) & 0x1FFFFFFull) | (2u << 30); // global_addr[56:32], type=2
    i32x8 g1;
    g1[0] = (2 << 16);                 // data_size = 4 bytes
    g1[1] = (int)(256u << 16);         // tensor_dim0[15:0] = 256
    g1[2] = (int)(1u << 16);           // tensor_dim0[31:16]=0, tensor_dim1=1
    g1[3] = (int)(256u << 16);         // tile_dim0 = 256 dwords (1KB)
    g1[4] = 0;                         // tile_dim1/2 unused (1-D)
    g1[5] = 256;                       // tensor_dim0_stride = 256
    g1[6] = 0;
    g1[7] = 0;
    i32x4 z4 = {0, 0, 0, 0};
    TDM_LOAD_TO_LDS(g0, g1, z4, z4);
    __builtin_amdgcn_s_wait_tensorcnt(0);
  }
  // level/valid words via plain loads (tiny)
  {
    int jidx = jt * 64 + t;
    jl[t] = (jidx < K_SEL) ? s_lv[(size_t)img * K_PAD + jidx] : 0u;
  }
  __syncthreads();

  const int i = it * 64 + t;
  float4 bi = make_float4(0.f, 0.f, 0.f, 0.f);
  u32 li = 0u;
  if (i < K_SEL) { bi = s_box[(size_t)img * K_PAD + i]; li = s_lv[(size_t)img * K_PAD + i]; }

  u64 w = 0ull;
  if (i < K_SEL && (li & 0x100u)) {
    float areai = (bi.z - bi.x) * (bi.w - bi.y);
    #pragma unroll 4
    for (int jj = 0; jj < 64; ++jj) {
      int j = jt * 64 + jj;
      if (j <= i || j >= K_SEL) continue;
      u32 lj = jl[jj];
      if (!(lj & 0x100u)) continue;
      if ((lj & 0xFFu) != (li & 0xFFu)) continue;   // cross-level: IoU treated as 0
      float4 bj = jb[jj];
      float ix1 = fmaxf(bi.x, bj.x), iy1 = fmaxf(bi.y, bj.y);
      float ix2 = fminf(bi.z, bj.z), iy2 = fminf(bi.w, bj.w);
      float iw = fmaxf(ix2 - ix1, 0.0f), ih = fmaxf(iy2 - iy1, 0.0f);
      float inter = iw * ih;
      float areaj = (bj.z - bj.x) * (bj.w - bj.y);
      float iou = inter / (areai + areaj - inter);
      if (iou > NMS_TH) w |= (1ull << jj);
    }
  }
  mask[((size_t)img * NMS_ROWS + i) * NMS_WORDS + jt] = w;
}

// =====================================================================
// Kernel 5: per-image sequential greedy reduction by one wave32 (lockstep,
// barrier-free), then ballot-based compaction of the first 1000 kept.
// 4 blocks x 256 threads.
// =====================================================================
__global__ void __launch_bounds__(256)
nms_reduce_kernel(const u64* __restrict__ mask, const float4* __restrict__ s_box,
                  const float* __restrict__ s_score, const u32* __restrict__ s_lv,
                  float* __restrict__ out) {
  const int img = blockIdx.x, tid = threadIdx.x;
  const size_t base = (size_t)img * K_PAD;

  __shared__ u64 removed[NMS_WORDS];
  __shared__ u32 wave_tot[8], wave_pref[8];
  __shared__ u32 s_total, s_chunk;

  // init removed: invalid entries and padding start removed
  for (int w = tid; w < NMS_WORDS; w += 256) {
    u64 r = 0ull;
    for (int bbit = 0; bbit < 64; ++bbit) {
      int p = w * 64 + bbit;
      if (p >= K_SEL || !(s_lv[base + p] & 0x100u)) r |= (1ull << bbit);
    }
    removed[w] = r;
  }
  if (tid == 0) s_total = 0u;
  __syncthreads();

  // --- sequential greedy pass: single wave32, no barriers needed ---
  if (tid < 32) {
    volatile u64* rm = removed;
    const int lane = tid;
    for (int i = 0; i < K_SEL; ++i) {
      const int wi = i >> 6;
      u64 cur = rm[wi];
      if (!((cur >> (i & 63)) & 1ull)) {
        const u64* row = mask + ((size_t)img * NMS_ROWS + i) * NMS_WORDS;
        for (int w = wi + lane; w < NMS_WORDS; w += 32) rm[w] |= row[w];
      }
      __builtin_amdgcn_wave_barrier();
    }
  }
  __syncthreads();

  // --- compact kept entries (in sorted score order) into output ---
  const int lane = tid & 31, wv = tid >> 5;
  for (int bstart = 0; bstart < K_SEL; bstart += 256) {
    int p = bstart + tid;
    bool kept = (p < K_SEL) && !((removed[p >> 6] >> (p & 63)) & 1ull);
    u64 bal = __ballot((int)kept);
    u32 prefix = (u32)__popcll(bal & ((1ull << lane) - 1ull));
    if (lane == 0) wave_tot[wv] = (u32)__popcll(bal);
    __syncthreads();
    if (tid == 0) {
      u32 acc = 0;
      for (int w = 0; w < 8; ++w) { wave_pref[w] = acc; acc += wave_tot[w]; }
      s_chunk = acc;
    }
    __syncthreads();
    u32 rank = s_total + wave_pref[wv] + prefix;
    if (kept && rank < POST_N) {
      float4 bb = s_box[base + p];
      float* ob = out + ((size_t)img * POST_N + rank) * 4;
      ob[0] = bb.x; ob[1] = bb.y; ob[2] = bb.z; ob[3] = bb.w;
      out[(size_t)BATCH * POST_N * 4 + (size_t)img * POST_N + rank] = s_score[base + p];
    }
    __syncthreads();
    if (tid == 0) s_total += s_chunk;
    __syncthreads();
  }

  // zero-pad remaining output slots (reference pads with zeros)
  u32 tot = s_total; if (tot > POST_N) tot = POST_N;
  for (u32 r = tot + tid; r < POST_N; r += 256) {
    float* ob = out + ((size_t)img * POST_N + r) * 4;
    ob[0] = 0.f; ob[1] = 0.f; ob[2] = 0.f; ob[3] = 0.f;
    out[(size_t)BATCH * POST_N * 4 + (size_t)img * POST_N + r] = 0.f;
  }
}

// =====================================================================
// Host-side launcher
// =====================================================================
extern "C" void kernel_launch(void* const* d_in, const int* in_sizes, int n_in,
                              void* d_out, int out_size, void* d_ws, size_t ws_size,
                              hipStream_t stream) {
  (void)in_sizes; (void)n_in; (void)out_size; (void)ws_size;
  const float* obj     = (const float*)d_in[0];   // [4, 159882]
  const float* deltas  = (const float*)d_in[1];   // [4, 159882, 4]
  const float* anchors = (const float*)d_in[2];   // [159882, 4]
  float* out = (float*)d_out;                     // 16000 boxes + 4000 scores

  char* w = (char*)d_ws;
  auto carve = [&](size_t bytes) -> void* {
    void* p = (void*)w;
    w += (bytes + 255) & ~(size_t)255;
    return p;
  };
  u32*    sel_key   = (u32*)   carve((size_t)BATCH * K_PAD * 4);
  int*    sel_idx   = (int*)   carve((size_t)BATCH * K_PAD * 4);
  float4* dec_box   = (float4*)carve((size_t)BATCH * K_PAD * 16);
  float*  dec_score = (float*) carve((size_t)BATCH * K_PAD * 4);
  u32*    dec_lvl   = (u32*)   carve((size_t)BATCH * K_PAD * 4);
  u32*    dec_key   = (u32*)   carve((size_t)BATCH * K_PAD * 4);
  float4* srt_box   = (float4*)carve((size_t)BATCH * K_PAD * 16);
  float*  srt_score = (float*) carve((size_t)BATCH * K_PAD * 4);
  u32*    srt_lv    = (u32*)   carve((size_t)BATCH * K_PAD * 4);
  u64*    supmask   = (u64*)   carve((size_t)BATCH * NMS_ROWS * NMS_WORDS * 8);

  topk_select_kernel<<<dim3(BATCH * 5), dim3(1024), 0, stream>>>(obj, sel_key, sel_idx);

  int decN = BATCH * K_SEL;
  decode_kernel<<<dim3((decN + 255) / 256), dim3(256), 0, stream>>>(
      obj, deltas, anchors, sel_idx, dec_box, dec_score, dec_lvl, dec_key);

  sort_kernel<<<dim3(BATCH), dim3(1024), PAD_N * sizeof(u64), stream>>>(
      dec_key, dec_box, dec_score, dec_lvl, srt_box, srt_score, srt_lv);

  nms_mask_kernel<<<dim3(NMS_WORDS, NMS_WORDS, BATCH), dim3(64), 0, stream>>>(
      srt_box, srt_lv, supmask);

  nms_reduce_kernel<<<dim3(BATCH), dim3(256), 0, stream>>>(
      supmask, srt_box, srt_score, srt_lv, out);
}